// GarNet_7962869367453
// MI455X (gfx1250) — compile-verified
//
#include <hip/hip_runtime.h>

typedef __attribute__((ext_vector_type(2))) float v2f;
typedef __attribute__((ext_vector_type(8))) float v8f;

// D = A(16x4 f32) x B(4x16 f32) + C(16x16 f32)
#define WMMA_F32(a, b, c) \
  __builtin_amdgcn_wmma_f32_16x16x4_f32(false, (a), false, (b), (short)0, (c), false, false)

__launch_bounds__(256)
__global__ void garnet_fused(const float* __restrict__ data,
                             const int*   __restrict__ num_vertex,
                             const float* __restrict__ W_flr,
                             const float* __restrict__ b_flr,
                             const float* __restrict__ W_s,
                             const float* __restrict__ b_s,
                             const float* __restrict__ W_out,
                             const float* __restrict__ b_out,
                             float*       __restrict__ out)
{
  constexpr int V = 1024, F = 16, P = 16, A = 8, NF = 32;

  __shared__ float ew_lds[V * A];          // 32 KB: edge weights ew[v][a]
  __shared__ float feat_stage[8][16 * 16]; //  8 KB: per-wave feature tile staging
  __shared__ float agg_part[8][16 * 16];   //  8 KB: per-wave partial aggregation
  __shared__ float agg_s[A * P];           // 512 B: reduced aggregation
  __shared__ float Mmat[A * NF];           //  1 KB: M[a][nf]

  const int b    = blockIdx.x;
  const int tid  = threadIdx.x;
  const int wave = tid >> 5;
  const int lane = tid & 31;
  const int lh   = lane >> 4;   // lane half (0/1)
  const int ln   = lane & 15;
  const int lnc  = ln & 7;      // clamped aggregator column (branch-free pad)
  const bool va  = (ln < A);    // valid aggregator lane

  const int nv = num_vertex[b];
  const float* dbase = data + (size_t)b * V * F;

  // ---- loop-invariant B fragments (B layout: lane ln = col N, elem r -> K) --
  v2f bwf[4], bws[4];
#pragma unroll
  for (int kc = 0; kc < 4; ++kc) {
    const int k = kc * 4 + lh * 2;
    bwf[kc][0] = W_flr[(k + 0) * P + ln];
    bwf[kc][1] = W_flr[(k + 1) * P + ln];
    // unconditional load at clamped column, then value-select (v_cndmask)
    const float w0 = W_s[(k + 0) * A + lnc];
    const float w1 = W_s[(k + 1) * A + lnc];
    bws[kc][0] = va ? w0 : 0.0f;
    bws[kc][1] = va ? w1 : 0.0f;
  }
  const float bf  = b_flr[ln];
  const float bsb = va ? b_s[lnc] : 0.0f;

  v8f aggC = {};  // persistent per-wave partial of ew^T @ feat

  // ---------------- Phase 1: features / edge weights / fused partial agg ----
  for (int t = 0; t < 8; ++t) {
    const int vbase = (wave * 8 + t) * 16;

    // A frags of the 16x16 data tile (A layout: lane ln = row M, elem r -> K)
    v2f af[4];
#pragma unroll
    for (int kc = 0; kc < 4; ++kc) {
      const int col = kc * 4 + lh * 2;
      af[kc] = *(const v2f*)(dbase + (size_t)(vbase + ln) * F + col);
    }
    if (t < 7) __builtin_prefetch(dbase + (size_t)(vbase + 16 + ln) * F, 0, 1);

    // features = data @ W_flr
    v8f feat = {};
#pragma unroll
    for (int kc = 0; kc < 4; ++kc) feat = WMMA_F32(af[kc], bwf[kc], feat);
#pragma unroll
    for (int i = 0; i < 8; ++i) feat[i] += bf;

    // distance = data @ W_s (cols >= 8 are zero-padded)
    v8f dist = {};
#pragma unroll
    for (int kc = 0; kc < 4; ++kc) dist = WMMA_F32(af[kc], bws[kc], dist);

    // edge weights: mask * 2^(-dist).  C row map: VGPR i -> v = vbase + i + lh*8
    // raw v_exp_f32 (single instruction) computed unconditionally, then select:
    // too cheap for the compiler to branchify -> pure v_cndmask masking.
    float e[8];
#pragma unroll
    for (int i = 0; i < 8; ++i) {
      const int v = vbase + i + lh * 8;
      const float ex = __builtin_amdgcn_exp2f(-(dist[i] + bsb));
      e[i] = (v < nv) ? ex : 0.0f;
      feat_stage[wave][(i + lh * 8) * 16 + ln] = feat[i];
    }
    if (va) {  // single exec toggle for all 8 ew stores
#pragma unroll
      for (int i = 0; i < 8; ++i)
        ew_lds[(vbase + i + lh * 8) * A + ln] = e[i];
    }

    // fused partial aggregation: aggC += ew_tile^T @ feat_tile (K = 16 verts)
    // A rows = aggregator a = ln (zero-padded for a >= 8), K = vertex
#pragma unroll
    for (int kc = 0; kc < 4; ++kc) {
      const int v0 = kc * 4 + lh * 2;
      v2f ae, bf2;
      const float e0 = ew_lds[(vbase + v0 + 0) * A + lnc];
      const float e1 = ew_lds[(vbase + v0 + 1) * A + lnc];
      ae[0]  = va ? e0 : 0.0f;
      ae[1]  = va ? e1 : 0.0f;
      bf2[0] = feat_stage[wave][(v0 + 0) * 16 + ln];
      bf2[1] = feat_stage[wave][(v0 + 1) * 16 + ln];
      aggC = WMMA_F32(ae, bf2, aggC);
    }
  }

  // stash per-wave partial aggregation (rows a >= 8 are exactly zero)
#pragma unroll
  for (int i = 0; i < 8; ++i)
    agg_part[wave][(i + lh * 8) * 16 + ln] = aggC[i];

  __syncthreads();

  // ---------------- Phase 2: reduce agg, build M ----------------------------
  if (tid < A * P) {
    const int a = tid >> 4, p = tid & 15;
    float s = 0.0f;
#pragma unroll
    for (int w = 0; w < 8; ++w) s += agg_part[w][a * 16 + p];
    agg_s[a * 16 + p] = s * (1.0f / (float)V);
  }
  __syncthreads();

  {
    const int a = tid >> 5, nf = tid & 31;  // 8 x 32 = 256 threads exactly
    float m = 0.0f;
#pragma unroll
    for (int p = 0; p < P; ++p)
      m += agg_s[a * 16 + p] * W_out[(a * 16 + p) * NF + nf];
    Mmat[a * NF + nf] = m;
  }
  __syncthreads();

  // ---------------- Phase 3: out = mask * (ew @ M + b_out) ------------------
  // Hoist M B-fragments and bias (invariant across tiles)
  v2f bm[2][2];
  float bo[2];
#pragma unroll
  for (int nt = 0; nt < 2; ++nt) {
    const int n0 = nt * 16;
    bo[nt] = b_out[n0 + ln];
#pragma unroll
    for (int kc = 0; kc < 2; ++kc) {
      const int k = kc * 4 + lh * 2;
      bm[nt][kc][0] = Mmat[(k + 0) * NF + n0 + ln];
      bm[nt][kc][1] = Mmat[(k + 1) * NF + n0 + ln];
    }
  }

  float* obase = out + (size_t)b * V * NF;
  for (int t = 0; t < 8; ++t) {
    const int vbase = (wave * 8 + t) * 16;

    // A frags from ew (K = A = 8 -> two chunks of 4); row = vertex
    v2f aw[2];
#pragma unroll
    for (int kc = 0; kc < 2; ++kc)
      aw[kc] = *(const v2f*)&ew_lds[(vbase + ln) * A + kc * 4 + lh * 2];

#pragma unroll
    for (int nt = 0; nt < 2; ++nt) {
      const int n0 = nt * 16;
      v8f oc = {bo[nt], bo[nt], bo[nt], bo[nt], bo[nt], bo[nt], bo[nt], bo[nt]};
#pragma unroll
      for (int kc = 0; kc < 2; ++kc) oc = WMMA_F32(aw[kc], bm[nt][kc], oc);
#pragma unroll
      for (int i = 0; i < 8; ++i) {
        const int v = vbase + i + lh * 8;
        obase[(size_t)v * NF + n0 + ln] = (v < nv) ? oc[i] : 0.0f;
      }
    }
  }
}

extern "C" void kernel_launch(void* const* d_in, const int* in_sizes, int n_in,
                              void* d_out, int out_size, void* d_ws, size_t ws_size,
                              hipStream_t stream) {
  const float* data       = (const float*)d_in[0];
  const int*   num_vertex = (const int*)d_in[1];
  const float* W_flr      = (const float*)d_in[2];
  const float* b_flr      = (const float*)d_in[3];
  const float* W_s        = (const float*)d_in[4];
  const float* b_s        = (const float*)d_in[5];
  const float* W_out      = (const float*)d_in[6];
  const float* b_out      = (const float*)d_in[7];
  float* out = (float*)d_out;

  garnet_fused<<<dim3(256), dim3(256), 0, stream>>>(
      data, num_vertex, W_flr, b_flr, W_s, b_s, W_out, b_out, out);
}